// AdaptiveScatteringNetwork_80582176407637
// MI455X (gfx1250) — compile-verified
//
#include <hip/hip_runtime.h>
#include <math.h>

typedef float v2f __attribute__((ext_vector_type(2)));
typedef float v8f __attribute__((ext_vector_type(8)));

#define SZ 128
#define PITCH 130
#define PLANE (SZ * PITCH) // floats per LDS plane

static __device__ __forceinline__ v8f vzero8() {
    v8f v = {0.f, 0.f, 0.f, 0.f, 0.f, 0.f, 0.f, 0.f};
    return v;
}

static __device__ __forceinline__ v8f wmma4(v2f a, v2f b, v8f c) {
    // D = A(16x4) * B(4x16) + C, f32
    return __builtin_amdgcn_wmma_f32_16x16x4_f32(false, a, false, b, (short)0, c, false, false);
}

// ---------------------------------------------------------------------------
// init: DFT matrix F[j][k] = exp(-2*pi*i*j*k/128), zero s-accumulators
// ---------------------------------------------------------------------------
__global__ void k_init(float* __restrict__ Fre, float* __restrict__ Fim,
                       float* __restrict__ acc, int accN) {
    int i = blockIdx.x * blockDim.x + threadIdx.x;
    if (i < SZ * SZ) {
        int r = i >> 7, c = i & 127;
        int t = (r * c) & 127; // j*k mod 128 (exact twiddle index)
        float ang = -6.283185307179586f * (float)t * (1.f / 128.f);
        Fre[i] = cosf(ang);
        Fim[i] = sinf(ang);
    }
    if (i < accN) acc[i] = 0.f;
}

// ---------------------------------------------------------------------------
// psi = mags * exp(i*phases)
// ---------------------------------------------------------------------------
__global__ void k_psi(const float* __restrict__ mags, const float* __restrict__ ph,
                      float* __restrict__ Pre, float* __restrict__ Pim, int n) {
    int i = blockIdx.x * blockDim.x + threadIdx.x;
    if (i < n) {
        float m = mags[i], p = ph[i];
        Pre[i] = m * cosf(p);
        Pim[i] = m * sinf(p);
    }
}

// ---------------------------------------------------------------------------
// xf = fft2(image[b]) = F @ img @ F   (and s0[b] = mean(img))
// one block per b; 8 waves, wave w owns tile-row w (rows 16w..16w+15)
// ---------------------------------------------------------------------------
__global__ void k_xf(const float* __restrict__ img,
                     const float* __restrict__ Fre, const float* __restrict__ Fim,
                     float* __restrict__ XFre, float* __restrict__ XFim,
                     float* __restrict__ S0) {
    extern __shared__ float sm[];
    float* TR = sm;
    float* TI = sm + PLANE;
    __shared__ float red[256];

    int b = blockIdx.x;
    const float* im0 = img + b * SZ * SZ;
    int tid = threadIdx.x;

    // s0
    float s = 0.f;
    for (int i = tid; i < SZ * SZ; i += 256) s += im0[i];
    red[tid] = s;
    __syncthreads();
    for (int o = 128; o > 0; o >>= 1) {
        if (tid < o) red[tid] += red[tid + o];
        __syncthreads();
    }
    if (tid == 0) S0[b] = red[0] * (1.f / 16384.f);

    int lane = tid & 31, wave = tid >> 5;
    int hs = lane >> 4, l16 = lane & 15;
    int tm = wave;

    // T = img @ F   (A real)
    for (int tn = 0; tn < 8; ++tn) {
        v8f RR = vzero8(), RI = vzero8();
        for (int kk = 0; kk < 32; ++kk) {
            int k0 = kk * 4;
            int rowA = tm * 16 + l16, kA = k0 + 2 * hs;
            v2f a;
            a.x = im0[rowA * SZ + kA];
            a.y = im0[rowA * SZ + kA + 1];
            int colB = tn * 16 + l16, rB = k0 + 2 * hs;
            v2f br, bi;
            br.x = Fre[rB * SZ + colB];
            br.y = Fre[(rB + 1) * SZ + colB];
            bi.x = Fim[rB * SZ + colB];
            bi.y = Fim[(rB + 1) * SZ + colB];
            RR = wmma4(a, br, RR);
            RI = wmma4(a, bi, RI);
        }
        for (int i = 0; i < 8; ++i) {
            int row = tm * 16 + 8 * hs + i, col = tn * 16 + l16;
            TR[row * PITCH + col] = RR[i];
            TI[row * PITCH + col] = RI[i];
        }
    }
    __syncthreads();

    // XF = F @ T
    float* xr = XFre + b * SZ * SZ;
    float* xi = XFim + b * SZ * SZ;
    for (int tn = 0; tn < 8; ++tn) {
        v8f RR = vzero8(), II = vzero8(), RI = vzero8(), IR = vzero8();
        for (int kk = 0; kk < 32; ++kk) {
            int k0 = kk * 4;
            int rowA = tm * 16 + l16, kA = k0 + 2 * hs;
            int i0 = rowA * SZ + kA;
            v2f ar, ai;
            ar.x = Fre[i0];
            ar.y = Fre[i0 + 1];
            ai.x = Fim[i0];
            ai.y = Fim[i0 + 1];
            int colB = tn * 16 + l16, rB = k0 + 2 * hs;
            v2f br, bi;
            br.x = TR[rB * PITCH + colB];
            br.y = TR[(rB + 1) * PITCH + colB];
            bi.x = TI[rB * PITCH + colB];
            bi.y = TI[(rB + 1) * PITCH + colB];
            RR = wmma4(ar, br, RR);
            II = wmma4(ai, bi, II);
            RI = wmma4(ar, bi, RI);
            IR = wmma4(ai, br, IR);
        }
        for (int i = 0; i < 8; ++i) {
            int row = tm * 16 + 8 * hs + i, col = tn * 16 + l16;
            xr[row * SZ + col] = RR[i] - II[i];
            xi[row * SZ + col] = RI[i] + IR[i];
        }
    }
}

// ---------------------------------------------------------------------------
// main scattering kernel: one block per (b, j1, l1)
// ---------------------------------------------------------------------------
__global__ void k_scatter(const float* __restrict__ Fre, const float* __restrict__ Fim,
                          const float* __restrict__ Pre, const float* __restrict__ Pim,
                          const float* __restrict__ XFre, const float* __restrict__ XFim,
                          float* __restrict__ S1, float* __restrict__ S2) {
    extern __shared__ float sm[];
    float* TR = sm;
    float* TI = sm + PLANE;
    float* WR = sm + 2 * PLANE;
    float* WI = sm + 3 * PLANE;

    int blk = blockIdx.x;
    int b = blk / 24;
    int rem = blk % 24;
    int j1 = rem >> 2;
    int l1 = rem & 3;

    const float* xr = XFre + b * SZ * SZ;
    const float* xi = XFim + b * SZ * SZ;
    const float* p1r = Pre + (j1 * 4 + l1) * SZ * SZ;
    const float* p1i = Pim + (j1 * 4 + l1) * SZ * SZ;

    int tid = threadIdx.x, lane = tid & 31, wave = tid >> 5;
    int hs = lane >> 4, l16 = lane & 15, tm = wave;
    const float inv2 = 1.f / 16384.f;

    // ---- GEMM1: T = (XF .* psi1) @ conj(F)   (psi-mult fused into A-frag load)
    for (int tn = 0; tn < 8; ++tn) {
        v8f RR = vzero8(), II = vzero8(), RI = vzero8(), IR = vzero8();
        for (int kk = 0; kk < 32; ++kk) {
            int k0 = kk * 4;
            int rowA = tm * 16 + l16, kA = k0 + 2 * hs;
            int i0 = rowA * SZ + kA;
            float x0r = xr[i0], x0i = xi[i0], q0r = p1r[i0], q0i = p1i[i0];
            float x1r = xr[i0 + 1], x1i = xi[i0 + 1], q1r = p1r[i0 + 1], q1i = p1i[i0 + 1];
            v2f ar, ai;
            ar.x = x0r * q0r - x0i * q0i;
            ai.x = x0r * q0i + x0i * q0r;
            ar.y = x1r * q1r - x1i * q1i;
            ai.y = x1r * q1i + x1i * q1r;
            int colB = tn * 16 + l16, rB = k0 + 2 * hs;
            v2f br, bi;
            br.x = Fre[rB * SZ + colB];
            br.y = Fre[(rB + 1) * SZ + colB];
            bi.x = Fim[rB * SZ + colB];
            bi.y = Fim[(rB + 1) * SZ + colB];
            RR = wmma4(ar, br, RR);
            II = wmma4(ai, bi, II);
            RI = wmma4(ar, bi, RI);
            IR = wmma4(ai, br, IR);
        }
        for (int i = 0; i < 8; ++i) {
            int row = tm * 16 + 8 * hs + i, col = tn * 16 + l16;
            TR[row * PITCH + col] = RR[i] + II[i]; // conj(B): re = RR + II
            TI[row * PITCH + col] = IR[i] - RI[i]; //          im = IR - RI
        }
    }
    __syncthreads();

    // ---- GEMM2: U = conj(F) @ T ; u1 = |U|/N^2 -> WR ; accumulate s1
    float accS1 = 0.f;
    for (int tn = 0; tn < 8; ++tn) {
        v8f RR = vzero8(), II = vzero8(), RI = vzero8(), IR = vzero8();
        for (int kk = 0; kk < 32; ++kk) {
            int k0 = kk * 4;
            int rowA = tm * 16 + l16, kA = k0 + 2 * hs;
            int i0 = rowA * SZ + kA;
            v2f ar, ai;
            ar.x = Fre[i0];
            ar.y = Fre[i0 + 1];
            ai.x = Fim[i0];
            ai.y = Fim[i0 + 1];
            int colB = tn * 16 + l16, rB = k0 + 2 * hs;
            v2f br, bi;
            br.x = TR[rB * PITCH + colB];
            br.y = TR[(rB + 1) * PITCH + colB];
            bi.x = TI[rB * PITCH + colB];
            bi.y = TI[(rB + 1) * PITCH + colB];
            RR = wmma4(ar, br, RR);
            II = wmma4(ai, bi, II);
            RI = wmma4(ar, bi, RI);
            IR = wmma4(ai, br, IR);
        }
        for (int i = 0; i < 8; ++i) {
            int row = tm * 16 + 8 * hs + i, col = tn * 16 + l16;
            float re = RR[i] + II[i]; // conj(A): re = RR + II
            float imv = RI[i] - IR[i]; //          im = RI - IR
            float u = sqrtf(re * re + imv * imv) * inv2;
            WR[row * PITCH + col] = u;
            accS1 += u;
        }
    }
    for (int o = 16; o > 0; o >>= 1) accS1 += __shfl_xor(accS1, o, 32);
    if (lane == 0) atomicAdd(&S1[b * 6 + j1], accS1 * (1.f / 65536.f));
    __syncthreads();

    // ---- GEMM3: T2 = u1 @ F   (A real from LDS)
    for (int tn = 0; tn < 8; ++tn) {
        v8f RR = vzero8(), RI = vzero8();
        for (int kk = 0; kk < 32; ++kk) {
            int k0 = kk * 4;
            int rowA = tm * 16 + l16, kA = k0 + 2 * hs;
            v2f a;
            a.x = WR[rowA * PITCH + kA];
            a.y = WR[rowA * PITCH + kA + 1];
            int colB = tn * 16 + l16, rB = k0 + 2 * hs;
            v2f br, bi;
            br.x = Fre[rB * SZ + colB];
            br.y = Fre[(rB + 1) * SZ + colB];
            bi.x = Fim[rB * SZ + colB];
            bi.y = Fim[(rB + 1) * SZ + colB];
            RR = wmma4(a, br, RR);
            RI = wmma4(a, bi, RI);
        }
        for (int i = 0; i < 8; ++i) {
            int row = tm * 16 + 8 * hs + i, col = tn * 16 + l16;
            TR[row * PITCH + col] = RR[i];
            TI[row * PITCH + col] = RI[i];
        }
    }
    __syncthreads();

    // ---- GEMM4: W = F @ T2  (= u1f, persists across second order)
    for (int tn = 0; tn < 8; ++tn) {
        v8f RR = vzero8(), II = vzero8(), RI = vzero8(), IR = vzero8();
        for (int kk = 0; kk < 32; ++kk) {
            int k0 = kk * 4;
            int rowA = tm * 16 + l16, kA = k0 + 2 * hs;
            int i0 = rowA * SZ + kA;
            v2f ar, ai;
            ar.x = Fre[i0];
            ar.y = Fre[i0 + 1];
            ai.x = Fim[i0];
            ai.y = Fim[i0 + 1];
            int colB = tn * 16 + l16, rB = k0 + 2 * hs;
            v2f br, bi;
            br.x = TR[rB * PITCH + colB];
            br.y = TR[(rB + 1) * PITCH + colB];
            bi.x = TI[rB * PITCH + colB];
            bi.y = TI[(rB + 1) * PITCH + colB];
            RR = wmma4(ar, br, RR);
            II = wmma4(ai, bi, II);
            RI = wmma4(ar, bi, RI);
            IR = wmma4(ai, br, IR);
        }
        for (int i = 0; i < 8; ++i) {
            int row = tm * 16 + 8 * hs + i, col = tn * 16 + l16;
            WR[row * PITCH + col] = RR[i] - II[i];
            WI[row * PITCH + col] = RI[i] + IR[i];
        }
    }
    __syncthreads();

    // ---- second order: for j2 > j1, l2: V = conj(F) @ ((W .* psi2) @ conj(F))
    int s2base = b * 15 + j1 * 5 - (j1 * (j1 - 1)) / 2; // offsets 0,5,9,12,14
    for (int j2 = j1 + 1; j2 < 6; ++j2) {
        float accS2 = 0.f;
        for (int l2 = 0; l2 < 4; ++l2) {
            const float* p2r = Pre + (j2 * 4 + l2) * SZ * SZ;
            const float* p2i = Pim + (j2 * 4 + l2) * SZ * SZ;

            // GEMM5: T3 = (W .* psi2) @ conj(F)
            for (int tn = 0; tn < 8; ++tn) {
                v8f RR = vzero8(), II = vzero8(), RI = vzero8(), IR = vzero8();
                for (int kk = 0; kk < 32; ++kk) {
                    int k0 = kk * 4;
                    int rowA = tm * 16 + l16, kA = k0 + 2 * hs;
                    int iL = rowA * PITCH + kA;
                    int iG = rowA * SZ + kA;
                    float w0r = WR[iL], w0i = WI[iL], q0r = p2r[iG], q0i = p2i[iG];
                    float w1r = WR[iL + 1], w1i = WI[iL + 1], q1r = p2r[iG + 1], q1i = p2i[iG + 1];
                    v2f ar, ai;
                    ar.x = w0r * q0r - w0i * q0i;
                    ai.x = w0r * q0i + w0i * q0r;
                    ar.y = w1r * q1r - w1i * q1i;
                    ai.y = w1r * q1i + w1i * q1r;
                    int colB = tn * 16 + l16, rB = k0 + 2 * hs;
                    v2f br, bi;
                    br.x = Fre[rB * SZ + colB];
                    br.y = Fre[(rB + 1) * SZ + colB];
                    bi.x = Fim[rB * SZ + colB];
                    bi.y = Fim[(rB + 1) * SZ + colB];
                    RR = wmma4(ar, br, RR);
                    II = wmma4(ai, bi, II);
                    RI = wmma4(ar, bi, RI);
                    IR = wmma4(ai, br, IR);
                }
                for (int i = 0; i < 8; ++i) {
                    int row = tm * 16 + 8 * hs + i, col = tn * 16 + l16;
                    TR[row * PITCH + col] = RR[i] + II[i];
                    TI[row * PITCH + col] = IR[i] - RI[i];
                }
            }
            __syncthreads();

            // GEMM6: V = conj(F) @ T3, |V|/N^2 reduced in registers (no store)
            for (int tn = 0; tn < 8; ++tn) {
                v8f RR = vzero8(), II = vzero8(), RI = vzero8(), IR = vzero8();
                for (int kk = 0; kk < 32; ++kk) {
                    int k0 = kk * 4;
                    int rowA = tm * 16 + l16, kA = k0 + 2 * hs;
                    int i0 = rowA * SZ + kA;
                    v2f ar, ai;
                    ar.x = Fre[i0];
                    ar.y = Fre[i0 + 1];
                    ai.x = Fim[i0];
                    ai.y = Fim[i0 + 1];
                    int colB = tn * 16 + l16, rB = k0 + 2 * hs;
                    v2f br, bi;
                    br.x = TR[rB * PITCH + colB];
                    br.y = TR[(rB + 1) * PITCH + colB];
                    bi.x = TI[rB * PITCH + colB];
                    bi.y = TI[(rB + 1) * PITCH + colB];
                    RR = wmma4(ar, br, RR);
                    II = wmma4(ai, bi, II);
                    RI = wmma4(ar, bi, RI);
                    IR = wmma4(ai, br, IR);
                }
                for (int i = 0; i < 8; ++i) {
                    float re = RR[i] + II[i];
                    float imv = RI[i] - IR[i];
                    accS2 += sqrtf(re * re + imv * imv) * inv2;
                }
            }
            __syncthreads();
        }
        for (int o = 16; o > 0; o >>= 1) accS2 += __shfl_xor(accS2, o, 32);
        if (lane == 0) atomicAdd(&S2[s2base + (j2 - j1 - 1)], accS2 * (1.f / 262144.f));
    }
}

// ---------------------------------------------------------------------------
// tiny MLP head: x(22) -> relu(16) -> relu(16) -> sigmoid(1)
// ---------------------------------------------------------------------------
__global__ void k_mlp(const float* __restrict__ S0, const float* __restrict__ S1v,
                      const float* __restrict__ S2v,
                      const float* __restrict__ w1, const float* __restrict__ b1,
                      const float* __restrict__ w2, const float* __restrict__ b2,
                      const float* __restrict__ w3, const float* __restrict__ b3,
                      float* __restrict__ out) {
    int b = threadIdx.x;
    if (b >= 64) return;
    float x[22];
    x[0] = S0[b];
    for (int j = 0; j < 6; ++j) x[1 + j] = S1v[b * 6 + j];
    for (int j = 0; j < 15; ++j) x[7 + j] = S2v[b * 15 + j];
    float h1[16];
    for (int j = 0; j < 16; ++j) {
        float a = b1[j];
        for (int i = 0; i < 22; ++i) a += x[i] * w1[i * 16 + j];
        h1[j] = fmaxf(a, 0.f);
    }
    float h2[16];
    for (int j = 0; j < 16; ++j) {
        float a = b2[j];
        for (int i = 0; i < 16; ++i) a += h1[i] * w2[i * 16 + j];
        h2[j] = fmaxf(a, 0.f);
    }
    float o = b3[0];
    for (int i = 0; i < 16; ++i) o += h2[i] * w3[i];
    out[b] = 1.f / (1.f + expf(-o));
}

// ---------------------------------------------------------------------------
extern "C" void kernel_launch(void* const* d_in, const int* in_sizes, int n_in,
                              void* d_out, int out_size, void* d_ws, size_t ws_size,
                              hipStream_t stream) {
    const float* img = (const float*)d_in[0];    // (64,128,128)
    const float* mags = (const float*)d_in[1];   // (6,4,128,128)
    const float* phases = (const float*)d_in[2]; // (6,4,128,128)
    const float* w1 = (const float*)d_in[3];
    const float* b1 = (const float*)d_in[4];
    const float* w2 = (const float*)d_in[5];
    const float* b2 = (const float*)d_in[6];
    const float* w3 = (const float*)d_in[7];
    const float* b3 = (const float*)d_in[8];
    float* out = (float*)d_out;

    float* ws = (float*)d_ws;
    float* Fre = ws;
    float* Fim = Fre + 16384;
    float* Pre = Fim + 16384;
    float* Pim = Pre + 393216;
    float* XFre = Pim + 393216;
    float* XFim = XFre + 64 * 16384;
    float* S0 = XFim + 64 * 16384;
    float* S1v = S0 + 64;
    float* S2v = S1v + 384;

    const int smXF = 2 * PLANE * 4;      // 133,120 B
    const int smSC = 4 * PLANE * 4;      // 266,240 B (<= 320KB WGP LDS)
    hipFuncSetAttribute((const void*)k_xf, hipFuncAttributeMaxDynamicSharedMemorySize, smXF);
    hipFuncSetAttribute((const void*)k_scatter, hipFuncAttributeMaxDynamicSharedMemorySize, smSC);

    k_init<<<64, 256, 0, stream>>>(Fre, Fim, S0, 64 + 384 + 960);
    k_psi<<<(393216 + 255) / 256, 256, 0, stream>>>(mags, phases, Pre, Pim, 393216);
    k_xf<<<64, 256, smXF, stream>>>(img, Fre, Fim, XFre, XFim, S0);
    k_scatter<<<64 * 24, 256, smSC, stream>>>(Fre, Fim, Pre, Pim, XFre, XFim, S1v, S2v);
    k_mlp<<<1, 64, 0, stream>>>(S0, S1v, S2v, w1, b1, w2, b2, w3, b3, out);
}